// SOMLayer_74062416052521
// MI455X (gfx1250) — compile-verified
//
#include <hip/hip_runtime.h>
#include <hip/hip_bf16.h>
#include <math.h>

typedef float v2f __attribute__((ext_vector_type(2)));
typedef float v8f __attribute__((ext_vector_type(8)));

#define NODES 4096
#define DDIM  1024
#define BATCH 8192

#define R2       (70.400001f * 70.400001f)                       // radius^2
#define INV2R2   (1.0f / (2.0f * 70.400001f * 70.400001f))
#define ALPHA    0.5f

// ---------------- K0: hn[n] = 0.5 * ||map_w[n]||^2 -------------------------
__global__ __launch_bounds__(256)
void k_halfnorm(const float* __restrict__ w, float* __restrict__ hn) {
    int wave = (blockIdx.x * blockDim.x + threadIdx.x) >> 5;   // one wave per node
    int lane = threadIdx.x & 31;
    const float4* row = (const float4*)(w + (size_t)wave * DDIM);
    float s = 0.f;
    #pragma unroll
    for (int i = 0; i < DDIM / 4 / 32; ++i) {
        float4 v = row[lane + 32 * i];
        s += v.x * v.x + v.y * v.y + v.z * v.z + v.w * v.w;
    }
    #pragma unroll
    for (int m = 16; m; m >>= 1) s += __shfl_xor(s, m, 32);
    if (lane == 0) hn[wave] = 0.5f * s;
}

// ---------------- K1: BMU search (fp32 WMMA GEMM + fused argmax) -----------
// block = 128 threads = 4 waves; each wave owns 16 batch rows; WG owns 64 rows.
// x rows staged in LDS with +4 float row pad (row stride 1028 -> bank stride 4).
// Each K-pass computes TWO 16-wide N-tiles sharing one A fragment (2x WMMA/load).
__global__ __launch_bounds__(128)
void k_bmu(const float* __restrict__ x, const float* __restrict__ w,
           const float* __restrict__ hn, int* __restrict__ bmu) {
    extern __shared__ float xs[];                 // [64][1028]
    const int ROWP = DDIM + 4;
    int tid  = threadIdx.x;
    int wid  = tid >> 5;
    int lane = tid & 31;
    int btile = blockIdx.x * 64;

    // cooperative stage of 64 x-rows
    for (int r = wid; r < 64; r += 4) {
        const float4* src = (const float4*)(x + (size_t)(btile + r) * DDIM);
        float4*       dst = (float4*)(xs + r * ROWP);
        for (int c = lane; c < DDIM / 4; c += 32) dst[c] = src[c];
    }
    __syncthreads();

    int mn  = lane & 15;            // A: M index / B: N index
    int hk  = (lane >> 4) * 2;      // K sub-offset (0 or 2)
    const float* arow = xs + (wid * 16 + mn) * ROWP + hk;

    float best0[8], best1[8];
    int   bidx0[8], bidx1[8];
    #pragma unroll
    for (int r = 0; r < 8; ++r) {
        best0[r] = -3.4e38f; bidx0[r] = 0;
        best1[r] = -3.4e38f; bidx1[r] = NODES - 1;
    }

    for (int nt = 0; nt < NODES; nt += 32) {
        int ncol0 = nt + mn;
        int ncol1 = nt + 16 + mn;
        float i0 = -hn[ncol0];                    // score = x·m - 0.5||m||^2
        float i1 = -hn[ncol1];
        v8f c0, c1;
        #pragma unroll
        for (int r = 0; r < 8; ++r) { c0[r] = i0; c1[r] = i1; }
        const float* b0p = w + (size_t)ncol0 * DDIM + hk;
        const float* b1p = w + (size_t)ncol1 * DDIM + hk;
        #pragma unroll 4
        for (int k = 0; k < DDIM; k += 4) {
            v2f a  = *(const v2f*)(arow + k);
            v2f b0 = *(const v2f*)(b0p + k);
            v2f b1 = *(const v2f*)(b1p + k);
            c0 = __builtin_amdgcn_wmma_f32_16x16x4_f32(
                     false, a, false, b0, (short)0, c0, false, false);
            c1 = __builtin_amdgcn_wmma_f32_16x16x4_f32(
                     false, a, false, b1, (short)0, c1, false, false);
        }
        #pragma unroll
        for (int r = 0; r < 8; ++r) {
            if (c0[r] > best0[r]) { best0[r] = c0[r]; bidx0[r] = ncol0; }
            if (c1[r] > best1[r]) { best1[r] = c1[r]; bidx1[r] = ncol1; }
        }
    }
    // merge the two tile streams (tile0 index < tile1 index within a pass,
    // but across passes indices interleave -> use index tie-break rule)
    float best[8]; int bidx[8];
    #pragma unroll
    for (int r = 0; r < 8; ++r) {
        bool take1 = (best1[r] > best0[r]) ||
                     (best1[r] == best0[r] && bidx1[r] < bidx0[r]);
        best[r] = take1 ? best1[r] : best0[r];
        bidx[r] = take1 ? bidx1[r] : bidx0[r];
    }

    // reduce across the 16 lanes holding one row (xor masks stay in-half)
    #pragma unroll
    for (int m = 8; m; m >>= 1) {
        #pragma unroll
        for (int r = 0; r < 8; ++r) {
            float ob = __shfl_xor(best[r], m, 32);
            int   oi = __shfl_xor(bidx[r], m, 32);
            if (ob > best[r] || (ob == best[r] && oi < bidx[r])) {
                best[r] = ob; bidx[r] = oi;
            }
        }
    }
    if (mn == 0) {
        int base = btile + wid * 16 + (lane >> 4) * 8;   // M = r + 8*(lane>>4)
        #pragma unroll
        for (int r = 0; r < 8; ++r) bmu[base + r] = bidx[r];
    }
}

// ---------------- K1b/K1c: histogram + sum_lr ------------------------------
__global__ void k_zero(int* __restrict__ cnt) {
    int i = blockIdx.x * blockDim.x + threadIdx.x;
    if (i < NODES) cnt[i] = 0;
}
__global__ void k_hist(const int* __restrict__ bmu, int* __restrict__ cnt) {
    int i = blockIdx.x * blockDim.x + threadIdx.x;
    if (i < BATCH) atomicAdd(&cnt[bmu[i]], 1);   // integer: deterministic
}
__global__ __launch_bounds__(256)
void k_sumlr(const int* __restrict__ cnt, float* __restrict__ sumlr) {
    int n    = (blockIdx.x * blockDim.x + threadIdx.x) >> 5;  // wave per node
    int lane = threadIdx.x & 31;
    int ni = n >> 6, nj = n & 63;
    float s = 0.f;
    for (int g = lane; g < NODES; g += 32) {
        int c = cnt[g];
        if (c) {
            float di = (float)((g >> 6) - ni);
            float dj = (float)((g & 63) - nj);
            float d2 = di * di + dj * dj;
            if (d2 <= R2) s += (float)c * __expf(-d2 * INV2R2);
        }
    }
    #pragma unroll
    for (int m = 16; m; m >>= 1) s += __shfl_xor(s, m, 32);
    if (lane == 0) sumlr[n] = ALPHA * s;
}

// ---------------- K2: wu = lr^T @ x via WMMA, fused final update -----------
// block = 128 = 4 waves. WG tile: 16 nodes x 256 d-cols; wave tile 16x64.
// A-fragment (lr) synthesized in registers from BMU indices each K-step.
// x K-chunks double-buffered in LDS via GLOBAL_LOAD_ASYNC_TO_LDS_B128.
__global__ __launch_bounds__(128)
void k_update(const float* __restrict__ x, const float* __restrict__ w,
              const int* __restrict__ bmu, const float* __restrict__ sumlr,
              float* __restrict__ out) {
    extern __shared__ char sm[];
    int*   bl  = (int*)sm;                          // [8192]        32 KB
    float* xs0 = (float*)(sm + BATCH * 4);          // [64][256]     64 KB
    float* xs1 = (float*)(sm + BATCH * 4 + 65536);  // [64][256]     64 KB

    int tid  = threadIdx.x;
    int wid  = tid >> 5;
    int lane = tid & 31;
    int ntile = (blockIdx.x >> 2) * 16;
    int dbase = (blockIdx.x & 3) * 256;
    int dloc  = wid * 64;                           // wave's 64 cols inside tile

    // async preload of BMU indices (ASYNCcnt-tracked, drained at loop top)
    {
        unsigned lb = (unsigned)(uintptr_t)bl;
        for (int i = tid * 4; i < BATCH; i += 128 * 4) {
            asm volatile("global_load_async_to_lds_b128 %0, %1, %2 offset:0"
                         :: "v"(lb + (unsigned)i * 4u),
                            "v"((unsigned)i * 4u),
                            "s"(bmu)
                         : "memory");
        }
    }

    int mn = lane & 15;
    int hk = (lane >> 4) * 2;
    int node = ntile + mn;                          // A-matrix row (M)
    int ni = node >> 6, nj = node & 63;
    int colb = dloc + mn;

    v8f acc[4];
    #pragma unroll
    for (int t = 0; t < 4; ++t) acc[t] = (v8f){};

    const int NCHUNK = BATCH / 64;                  // 128

    // issue one chunk's async loads: x[k0..k0+63][dbase..dbase+255] -> buf
    auto issue_chunk = [&](int chunk, float* buf) {
        int k0 = chunk * 64;
        for (int r = wid; r < 64; r += 4) {
            unsigned gbase = (unsigned)(((unsigned)(k0 + r) * DDIM + dbase) * 4u);
            unsigned lbase = (unsigned)(uintptr_t)(buf + r * 256);
            #pragma unroll
            for (int c = 0; c < 2; ++c) {
                unsigned off = (unsigned)(lane + 32 * c) * 16u;
                asm volatile("global_load_async_to_lds_b128 %0, %1, %2 offset:0"
                             :: "v"(lbase + off), "v"(gbase + off), "s"(x)
                             : "memory");
            }
        }
    };

    issue_chunk(0, xs0);

    for (int chunk = 0; chunk < NCHUNK; ++chunk) {
        // drain this wave's asyncs, then publish to whole WG
        asm volatile("s_wait_asynccnt 0x0" ::: "memory");
        __syncthreads();
        float* cur = (chunk & 1) ? xs1 : xs0;
        if (chunk + 1 < NCHUNK) issue_chunk(chunk + 1, (chunk & 1) ? xs0 : xs1);

        int k0 = chunk * 64;
        #pragma unroll 2
        for (int kk = 0; kk < 64; kk += 4) {
            int b0 = kk + hk;
            int g0 = bl[k0 + b0];
            int g1 = bl[k0 + b0 + 1];
            float di0 = (float)((g0 >> 6) - ni), dj0 = (float)((g0 & 63) - nj);
            float di1 = (float)((g1 >> 6) - ni), dj1 = (float)((g1 & 63) - nj);
            float d20 = di0 * di0 + dj0 * dj0;
            float d21 = di1 * di1 + dj1 * dj1;
            v2f a;
            a.x = (d20 <= R2) ? ALPHA * __expf(-d20 * INV2R2) : 0.f;
            a.y = (d21 <= R2) ? ALPHA * __expf(-d21 * INV2R2) : 0.f;
            const float* r0 = cur + b0 * 256 + colb;
            #pragma unroll
            for (int t = 0; t < 4; ++t) {
                v2f b;
                b.x = r0[16 * t];
                b.y = r0[256 + 16 * t];
                acc[t] = __builtin_amdgcn_wmma_f32_16x16x4_f32(
                             false, a, false, b, (short)0, acc[t], false, false);
            }
        }
        __syncthreads();
    }

    // fused: out = map + (wu - sumlr*map) / BATCH
    int mrow = 8 * (lane >> 4);
    #pragma unroll
    for (int r = 0; r < 8; ++r) {
        int n = ntile + mrow + r;
        float sl = sumlr[n];
        size_t rowoff = (size_t)n * DDIM;
        #pragma unroll
        for (int t = 0; t < 4; ++t) {
            int col = dbase + dloc + 16 * t + mn;
            float mw = w[rowoff + col];
            out[rowoff + col] = mw + (acc[t][r] - sl * mw) * (1.0f / 8192.0f);
        }
    }
}

// ---------------------------------------------------------------------------
extern "C" void kernel_launch(void* const* d_in, const int* in_sizes, int n_in,
                              void* d_out, int out_size, void* d_ws, size_t ws_size,
                              hipStream_t stream) {
    (void)in_sizes; (void)n_in; (void)out_size; (void)ws_size;
    const float* x = (const float*)d_in[0];
    const float* w = (const float*)d_in[1];
    // d_in[2] (location_vects) not needed: node (i,j) = (n>>6, n&63) by construction
    float* out = (float*)d_out;

    char* ws = (char*)d_ws;
    float* hn    = (float*)(ws);             // 16 KB
    float* sumlr = (float*)(ws + 16384);     // 16 KB
    int*   bmu   = (int*)  (ws + 32768);     // 32 KB
    int*   cnt   = (int*)  (ws + 65536);     // 16 KB

    k_halfnorm<<<512, 256, 0, stream>>>(w, hn);
    k_bmu<<<BATCH / 64, 128, 64 * (DDIM + 4) * sizeof(float), stream>>>(x, w, hn, bmu);
    k_zero<<<16, 256, 0, stream>>>(cnt);
    k_hist<<<BATCH / 256, 256, 0, stream>>>(bmu, cnt);
    k_sumlr<<<512, 256, 0, stream>>>(cnt, sumlr);
    k_update<<<(NODES / 16) * 4, 128,
               BATCH * 4 + 2 * 64 * 256 * 4, stream>>>(x, w, bmu, sumlr, out);
}